// EntropyConstrainedRefinement_67181878444997
// MI455X (gfx1250) — compile-verified
//
#include <hip/hip_runtime.h>

#define EPS_F 1e-10f
#define NWAVES 8   // 256 threads / wave32

// ---------------- wave32 / block reductions (deterministic order) ----------------
__device__ __forceinline__ float wave_sum(float v) {
#pragma unroll
  for (int o = 16; o > 0; o >>= 1) v += __shfl_down(v, o, 32);
  return v;
}
__device__ __forceinline__ float wave_max(float v) {
#pragma unroll
  for (int o = 16; o > 0; o >>= 1) v = fmaxf(v, __shfl_down(v, o, 32));
  return v;
}
__device__ __forceinline__ float wave_min(float v) {
#pragma unroll
  for (int o = 16; o > 0; o >>= 1) v = fminf(v, __shfl_down(v, o, 32));
  return v;
}

__device__ __forceinline__ float block_sum(float v, float* red) {
  v = wave_sum(v);
  const int lane = threadIdx.x & 31, wid = threadIdx.x >> 5;
  __syncthreads();
  if (lane == 0) red[wid] = v;
  __syncthreads();
  float r = red[0];
#pragma unroll
  for (int i = 1; i < NWAVES; ++i) r += red[i];
  return r;  // identical in every lane
}
__device__ __forceinline__ float block_max(float v, float* red) {
  v = wave_max(v);
  const int lane = threadIdx.x & 31, wid = threadIdx.x >> 5;
  __syncthreads();
  if (lane == 0) red[wid] = v;
  __syncthreads();
  float r = red[0];
#pragma unroll
  for (int i = 1; i < NWAVES; ++i) r = fmaxf(r, red[i]);
  return r;
}
__device__ __forceinline__ float block_min(float v, float* red) {
  v = wave_min(v);
  const int lane = threadIdx.x & 31, wid = threadIdx.x >> 5;
  __syncthreads();
  if (lane == 0) red[wid] = v;
  __syncthreads();
  float r = red[0];
#pragma unroll
  for (int i = 1; i < NWAVES; ++i) r = fminf(r, red[i]);
  return r;
}

// ---------------- Stage 1: entropy-constrained Frank-Wolfe, one block per row ----
__global__ __launch_bounds__(256) void fw_kernel(const float* __restrict__ q,
                                                 float* __restrict__ p_out,
                                                 int K) {
  __shared__ float red[NWAVES];
  const int b = blockIdx.x;
  const int tid = threadIdx.x;

  float qv[4];
  bool val[4];
#pragma unroll
  for (int j = 0; j < 4; ++j) {
    int i = tid + j * 256;
    val[j] = (i < K);
    qv[j] = val[j] ? q[(size_t)b * K + i] : 0.0f;
  }

  // row max (for stable softmax; identical to jax softmax shift)
  float m = -__builtin_inff();
#pragma unroll
  for (int j = 0; j < 4; ++j)
    if (val[j]) m = fmaxf(m, qv[j]);
  const float qmax = block_max(m, red);

  // ---- _solve_dual: 20-step binary search on lambda ----
  float lmin = 0.001f, lmax = 100.0f;
  float e[4];
  for (int it = 0; it < 20; ++it) {
    const float lmid = 0.5f * (lmin + lmax);
    float ls = 0.0f;
#pragma unroll
    for (int j = 0; j < 4; ++j) {
      e[j] = val[j] ? expf((qv[j] - qmax) / lmid) : 0.0f;
      ls += e[j];
    }
    const float Z = block_sum(ls, red);
    float t[4], st = 0.0f;
#pragma unroll
    for (int j = 0; j < 4; ++j) {
      t[j] = val[j] ? (e[j] / Z + EPS_F) : 0.0f;
      st += t[j];
    }
    const float St = block_sum(st, red);
    float le = 0.0f;
#pragma unroll
    for (int j = 0; j < 4; ++j)
      if (val[j]) {
        const float ps = t[j] / St;
        le += ps * logf(ps);
      }
    const float ent = -block_sum(le, red);
    const bool below = (ent < 0.6f);  // ENTROPY_THRESHOLD
    lmin = below ? lmid : lmin;
    lmax = below ? lmax : lmid;
  }

  // s = softmax(q / lmax)
  float ls = 0.0f;
#pragma unroll
  for (int j = 0; j < 4; ++j) {
    e[j] = val[j] ? expf((qv[j] - qmax) / lmax) : 0.0f;
    ls += e[j];
  }
  const float Z = block_sum(ls, red);
  float s[4];
#pragma unroll
  for (int j = 0; j < 4; ++j) s[j] = e[j] / Z;

  float acc = 0.0f;
#pragma unroll
  for (int j = 0; j < 4; ++j) acc += s[j] * qv[j];
  const float sq = block_sum(acc, red);

  // p0 uniform
  float p[4];
#pragma unroll
  for (int j = 0; j < 4; ++j) p[j] = val[j] ? (1.0f / 1000.0f) : 0.0f;

  // ---- Frank-Wolfe, 50 iterations (scalar decisions uniform across block) ----
  bool done = false;
  for (int it = 0; it < 50; ++it) {
    float pa = 0.0f;
#pragma unroll
    for (int j = 0; j < 4; ++j) pa += p[j] * qv[j];
    const float pq = block_sum(pa, red);

    // line search on linear objective: obj(g) = (1-g)*pq + g*sq
    float g[11];
#pragma unroll
    for (int jj = 0; jj < 11; ++jj) g[jj] = (jj == 10) ? 1.0f : 0.1f * (float)jj;
    int best = 0;
    float bv = -__builtin_inff();
#pragma unroll
    for (int jj = 0; jj < 11; ++jj) {
      const float o = (1.0f - g[jj]) * pq + g[jj] * sq;
      if (o > bv) { bv = o; best = jj; }  // first-max tie-break like jnp.argmax
    }
    const int lo = (best - 1 < 0) ? 0 : best - 1;
    const int hi = (best + 1 > 10) ? 10 : best + 1;
    float gmin = g[lo], gmax = g[hi];
#pragma unroll
    for (int t5 = 0; t5 < 5; ++t5) {
      const float gmid = 0.5f * (gmin + gmax);
      const float omin = (1.0f - gmin) * pq + gmin * sq;
      const float omax = (1.0f - gmax) * pq + gmax * sq;
      const float omid = (1.0f - gmid) * pq + gmid * sq;
      const bool peak = omid > fmaxf(omin, omax);
      const bool left = omin > omax;
      const float nmin = peak ? 0.5f * (gmin + gmid) : (left ? gmin : gmid);
      const float nmax = peak ? 0.5f * (gmid + gmax) : (left ? gmid : gmax);
      gmin = nmin; gmax = nmax;
    }
    const float gamma =
        (best == 0) ? 0.0f : ((best == 10) ? 1.0f : 0.5f * (gmin + gmax));

    float pn[4], ia = 0.0f;
#pragma unroll
    for (int j = 0; j < 4; ++j) {
      pn[j] = (1.0f - gamma) * p[j] + gamma * s[j];
      ia += (pn[j] - p[j]) * qv[j];
    }
    const float imp = block_sum(ia, red);
    const bool done_now = fabsf(imp) < 1e-4f;  // CONVERGENCE_TOL
    const bool take = (!done) && (!done_now);
    if (take) {
#pragma unroll
      for (int j = 0; j < 4; ++j) p[j] = pn[j];
    }
    done = done || done_now;
  }

#pragma unroll
  for (int j = 0; j < 4; ++j)
    if (val[j]) p_out[(size_t)b * K + tid + j * 256] = p[j];
}

// ---------------- Stage 2: total variation per 64x64 tile, async-LDS staged -----
__global__ __launch_bounds__(256) void tv_kernel(const float* __restrict__ masks,
                                                 float* __restrict__ tv) {
  __shared__ float sm[64 * 65];  // 65-float row stride: bank spreading
  __shared__ float red[NWAVES];
  const int tile = blockIdx.x;
  const int tid = threadIdx.x;

  const float* gbase = masks + (size_t)tile * 4096;
  const unsigned lbase = (unsigned)(uintptr_t)(&sm[0]);

  // CDNA5 async global->LDS: 4 x b128 per thread = 16KB tile, tracked on ASYNCcnt
#pragma unroll
  for (int j = 0; j < 4; ++j) {
    const int cidx = tid + j * 256;           // 16-byte chunk index, 0..1023
    const int row = cidx >> 4;                // 16 chunks per 64-float row
    const int cw = cidx & 15;
    const unsigned lofs = lbase + (unsigned)(row * 65 + cw * 4) * 4u;
    const unsigned gofs = (unsigned)cidx * 16u;
    asm volatile("global_load_async_to_lds_b128 %0, %1, %2"
                 :
                 : "v"(lofs), "v"(gofs), "s"(gbase)
                 : "memory");
  }
  asm volatile("s_wait_asynccnt 0" ::: "memory");
  __syncthreads();

  // thread -> (row r, 16-wide column strip)
  const int r = tid >> 2;
  const int c0 = (tid & 3) * 16;
  const float* rowp = &sm[r * 65];
  float acc = 0.0f;
#pragma unroll
  for (int c = 0; c < 16; ++c) {
    const int cc = c0 + c;
    const float a = rowp[cc];
    if (cc < 63) acc += fabsf(rowp[cc + 1] - a);   // h_grad
    if (r < 63) acc += fabsf(rowp[65 + cc] - a);   // v_grad
  }
  const float tot = block_sum(acc, red);
  if (tid == 0) tv[tile] = tot / 4096.0f;  // (H*W + 1e-10) rounds to 4096.0f
}

// ---------------- Stage 3: per-row normalize/blend --------------------------------
__global__ __launch_bounds__(256) void finalize_kernel(const float* __restrict__ p,
                                                       const float* __restrict__ tv,
                                                       float* __restrict__ out,
                                                       int K) {
  __shared__ float red[NWAVES];
  const int b = blockIdx.x;
  const int tid = threadIdx.x;

  float tvv[4], pv[4];
  bool val[4];
#pragma unroll
  for (int j = 0; j < 4; ++j) {
    int i = tid + j * 256;
    val[j] = (i < K);
    tvv[j] = val[j] ? tv[(size_t)b * K + i] : 0.0f;
    pv[j] = val[j] ? p[(size_t)b * K + i] : 0.0f;
  }
  float mn = __builtin_inff(), mx = -__builtin_inff();
#pragma unroll
  for (int j = 0; j < 4; ++j)
    if (val[j]) { mn = fminf(mn, tvv[j]); mx = fmaxf(mx, tvv[j]); }
  const float tmin = block_min(mn, red);
  const float tmax = block_max(mx, red);
  const float den = (tmax - tmin) + EPS_F;

  float pr[4], ss = 0.0f;
#pragma unroll
  for (int j = 0; j < 4; ++j) {
    const float w = 1.0f - (tvv[j] - tmin) / den;
    pr[j] = 0.9f * pv[j] + 0.1f * w;  // (1-LAMBDA_REG)*p + LAMBDA_REG*tv_weight
    if (val[j]) ss += pr[j];
  }
  const float S = block_sum(ss, red);
#pragma unroll
  for (int j = 0; j < 4; ++j)
    if (val[j]) out[(size_t)b * K + tid + j * 256] = pr[j] / S;
}

// ---------------- launch -----------------------------------------------------------
extern "C" void kernel_launch(void* const* d_in, const int* in_sizes, int n_in,
                              void* d_out, int out_size, void* d_ws, size_t ws_size,
                              hipStream_t stream) {
  // inputs: [0] scores (unused in forward), [1] quality_scores, [2] masks
  const float* q = (const float*)d_in[1];
  const float* masks = (const float*)d_in[2];
  float* out = (float*)d_out;

  const int K = 1000;
  const int B = in_sizes[1] / K;                 // 16
  const int n_tiles = in_sizes[2] / (64 * 64);   // B*K = 16000

  float* p_ws = out;              // stage FW result in d_out (finalize reads before writing)
  float* tv_ws = (float*)d_ws;    // n_tiles floats of scratch

  fw_kernel<<<dim3(B), dim3(256), 0, stream>>>(q, p_ws, K);
  tv_kernel<<<dim3(n_tiles), dim3(256), 0, stream>>>(masks, tv_ws);
  finalize_kernel<<<dim3(B), dim3(256), 0, stream>>>(p_ws, tv_ws, out, K);
}